// CapsuleLayer_60782377173364
// MI455X (gfx1250) — compile-verified
//
#include <hip/hip_runtime.h>
#include <hip/hip_bf16.h>

// CapsuleLayer dynamic routing, fully fused for gfx1250 (MI455X).
// One workgroup per (capsule c, batch tile of 16). Priors recomputed per pass
// with V_WMMA_F32_16X16X4_F32. Per-tile routing state lives in LDS:
//   logits [16][1152] (72KB)  +  attn (exp-precomputed) [1152][16] (72KB)
// -- only feasible thanks to CDNA5's 320KB/WGP LDS.

typedef __attribute__((ext_vector_type(2))) float v2f;
typedef __attribute__((ext_vector_type(8))) float v8f;

#define C_    10
#define B_    128
#define N_    1152
#define IN_   8
#define OUT_  16
#define ITERS 3
#define BT    16            // batch tile (WMMA M)
#define WAVES 8
#define THREADS (WAVES * 32)

__global__ __launch_bounds__(THREADS)
void caps_routing_kernel(const float* __restrict__ x,      // [B,N,IN]
                         const float* __restrict__ W,      // [C,N,IN,OUT]
                         float* __restrict__ out)          // [C,B,OUT]
{
    __shared__ float logits[BT][N_];     // 73728 B, collapsed over OUT (broadcast axis)
    __shared__ float attn[N_][BT];       // 73728 B, exp(logit - mx); [n][b] for b128 reads
    __shared__ float sArr[BT][OUT_];     // s = sum_n attn * priors (pre-normalization)
    __shared__ float oT[OUT_][BT];       // squash(s), transposed for pass-2 b128 reads
    __shared__ float red[BT][16];        // softmax reduction scratch
    __shared__ float mx[BT];             // per-row max
    __shared__ float rs[BT];             // 1 / sum(exp)

    const int c    = blockIdx.y;
    const int b0   = blockIdx.x * BT;
    const int tid  = threadIdx.x;
    const int lane = tid & 31;
    const int wave = tid >> 5;
    const int half = lane >> 4;          // 0: lanes 0-15, 1: lanes 16-31
    const int m    = lane & 15;          // A row (batch) / B-D col (out)

    // ---- init logits to zero ----
    for (int idx = tid; idx < BT * N_; idx += THREADS)
        (&logits[0][0])[idx] = 0.0f;
    __syncthreads();

    // Per-lane base pointers for WMMA operand gathers (constant over n / iter).
    // A (16x4 f32): lane holds x[b0+m][n][k0], x[b0+m][n][k0+1], k0 = kbase + 2*half
    // B (4x16 f32): lane holds W[c][n][k0][m],  W[c][n][k0+1][m]
    const float* xrow  = x + (size_t)(b0 + m) * (N_ * IN_);
    const float* wbase = W + (size_t)c * ((size_t)N_ * IN_ * OUT_);
    const int k0 = 2 * half;

    for (int it = 0; it < ITERS; ++it) {
        // ============ softmax over N per batch row; precompute attn ============
        {
            const int b = tid >> 4;      // 0..15
            const int j = tid & 15;      // 0..15
            float pm = -3.402823e38f;
            for (int n = j; n < N_; n += 16)
                pm = fmaxf(pm, logits[b][n]);
            red[b][j] = pm;
            __syncthreads();
            if (j == 0) {
                float v = red[b][0];
                #pragma unroll
                for (int k = 1; k < 16; ++k) v = fmaxf(v, red[b][k]);
                mx[b] = v;
            }
            __syncthreads();
            float ps = 0.0f;
            const float mb = mx[b];
            for (int n = j; n < N_; n += 16) {
                const float e = __expf(logits[b][n] - mb);
                attn[n][b] = e;          // unnormalized; 1/sum folded in after pass 1
                ps += e;
            }
            red[b][j] = ps;
            __syncthreads();
            if (j == 0) {
                float v = 0.0f;
                #pragma unroll
                for (int k = 0; k < 16; ++k) v += red[b][k];
                rs[b] = 1.0f / v;
            }
            if (tid < BT * OUT_) (&sArr[0][0])[tid] = 0.0f;
            __syncthreads();
        }

        // ========== pass 1: s[b,o] = sum_n attn[b,n] * priors[b,n,o] ==========
        v8f sacc = {};
        for (int n = wave; n < N_; n += WAVES) {   // uniform trip count: EXEC all-ones
            const float* xp = xrow + n * IN_;
            const float* wp = wbase + (size_t)n * (IN_ * OUT_);
            v2f a0, a1, bb0, bb1;
            a0.x  = xp[k0];                  a0.y  = xp[k0 + 1];
            a1.x  = xp[4 + k0];              a1.y  = xp[4 + k0 + 1];
            bb0.x = wp[(k0)     * OUT_ + m]; bb0.y = wp[(k0 + 1) * OUT_ + m];
            bb1.x = wp[(4 + k0) * OUT_ + m]; bb1.y = wp[(4 + k0 + 1) * OUT_ + m];
            v8f p = {};
            p = __builtin_amdgcn_wmma_f32_16x16x4_f32(false, a0, false, bb0,
                                                      (short)0, p, false, false);
            p = __builtin_amdgcn_wmma_f32_16x16x4_f32(false, a1, false, bb1,
                                                      (short)0, p, false, false);
            const float* ap = &attn[n][half * 8];  // 8 contiguous floats, 32B aligned
            #pragma unroll
            for (int r = 0; r < 8; ++r)
                sacc[r] += ap[r] * p[r];
        }
        #pragma unroll
        for (int r = 0; r < 8; ++r)
            atomicAdd(&sArr[r + half * 8][m], sacc[r] * rs[r + half * 8]);
        __syncthreads();

        // ====================== squash + (final) output ======================
        if (tid < BT) {
            const int b = tid;
            float sq = 0.0f;
            #pragma unroll
            for (int o = 0; o < OUT_; ++o) { const float v = sArr[b][o]; sq += v * v; }
            const float scale = (sq / (1.0f + sq)) * rsqrtf(sq + 1e-8f);
            #pragma unroll
            for (int o = 0; o < OUT_; ++o) {
                const float v = sArr[b][o] * scale;
                oT[o][b] = v;
                if (it == ITERS - 1)
                    out[((size_t)c * B_ + (b0 + b)) * OUT_ + o] = v;
            }
        }
        __syncthreads();

        // ==== pass 2: logits[b,n] += sum_o out[b,o] * priors[b,n,o] ====
        if (it < ITERS - 1) {
            for (int n = wave; n < N_; n += WAVES) {
                const float* xp = xrow + n * IN_;
                const float* wp = wbase + (size_t)n * (IN_ * OUT_);
                v2f a0, a1, bb0, bb1;
                a0.x  = xp[k0];                  a0.y  = xp[k0 + 1];
                a1.x  = xp[4 + k0];              a1.y  = xp[4 + k0 + 1];
                bb0.x = wp[(k0)     * OUT_ + m]; bb0.y = wp[(k0 + 1) * OUT_ + m];
                bb1.x = wp[(4 + k0) * OUT_ + m]; bb1.y = wp[(4 + k0 + 1) * OUT_ + m];
                v8f p = {};
                p = __builtin_amdgcn_wmma_f32_16x16x4_f32(false, a0, false, bb0,
                                                          (short)0, p, false, false);
                p = __builtin_amdgcn_wmma_f32_16x16x4_f32(false, a1, false, bb1,
                                                          (short)0, p, false, false);
                const float* op = &oT[m][half * 8]; // 8 contiguous floats per lane
                float t[8];
                #pragma unroll
                for (int r = 0; r < 8; ++r)
                    t[r] = op[r] * p[r];
                // reduce over the 16 'out' lanes (stays within each 16-lane half)
                #pragma unroll
                for (int off = 1; off < 16; off <<= 1) {
                    #pragma unroll
                    for (int r = 0; r < 8; ++r)
                        t[r] += __shfl_xor(t[r], off, 32);
                }
                if (m == 0) {
                    #pragma unroll
                    for (int r = 0; r < 8; ++r)
                        logits[r + half * 8][n] += t[r];
                }
            }
            __syncthreads();
        }
    }
}

extern "C" void kernel_launch(void* const* d_in, const int* in_sizes, int n_in,
                              void* d_out, int out_size, void* d_ws, size_t ws_size,
                              hipStream_t stream) {
    (void)in_sizes; (void)n_in; (void)out_size; (void)d_ws; (void)ws_size;
    const float* x = (const float*)d_in[0];          // [B,N,IN]
    const float* W = (const float*)d_in[1];          // [C,N,IN,OUT]
    float* out     = (float*)d_out;                  // [C,B,OUT]
    dim3 grid(B_ / BT, C_);
    dim3 block(THREADS);
    caps_routing_kernel<<<grid, block, 0, stream>>>(x, W, out);
}